// CanvasLayer_58471684768213
// MI455X (gfx1250) — compile-verified
//
#include <hip/hip_runtime.h>
#include <math.h>

// Problem constants (match reference setup_inputs)
#define B_   8
#define K_   4096
#define D_   192
#define ROWS (B_ * K_)      // 32768
#define KR   384            // concat read dim (2*D)

typedef __attribute__((ext_vector_type(2))) float v2f;
typedef __attribute__((ext_vector_type(8))) float v8f;

__device__ __forceinline__ float wave_sum(float v) {
    // wave32 all-reduce
    #pragma unroll
    for (int m = 16; m >= 1; m >>= 1) v += __shfl_xor(v, m, 32);
    return v;
}

// ---------------------------------------------------------------------------
// Prep: per-row layernorm -> gate scalar, bilinear gather of both canvases
// into R[row][0..383], and flat scatter indices. One wave per (b,k) row.
// ---------------------------------------------------------------------------
__device__ __forceinline__ void bilin6(const float* __restrict__ canvas, int b, int HWd,
                                       float gx, float gy, int lane, float out6[6]) {
    float x = (gx + 1.f) * 0.5f * (float)(HWd - 1);
    float y = (gy + 1.f) * 0.5f * (float)(HWd - 1);
    float x0f = floorf(x), y0f = floorf(y);
    float wx = x - x0f, wy = y - y0f;
    int x0 = min(max((int)x0f, 0), HWd - 1);
    int x1 = min(x0 + 1, HWd - 1);
    int y0 = min(max((int)y0f, 0), HWd - 1);
    int y1 = min(y0 + 1, HWd - 1);
    float w00 = (1.f - wx) * (1.f - wy), w01 = wx * (1.f - wy);
    float w10 = (1.f - wx) * wy,         w11 = wx * wy;
    size_t base = (size_t)b * D_ * HWd * HWd;
    #pragma unroll
    for (int j = 0; j < 6; ++j) {
        int ch = lane + 32 * j;
        const float* cp = canvas + base + (size_t)ch * HWd * HWd;
        float v00 = cp[y0 * HWd + x0];
        float v01 = cp[y0 * HWd + x1];
        float v10 = cp[y1 * HWd + x0];
        float v11 = cp[y1 * HWd + x1];
        out6[j] = v00 * w00 + v01 * w01 + v10 * w10 + v11 * w11;
    }
}

__global__ void prep_kernel(const float* __restrict__ h,
                            const float* __restrict__ c2,
                            const float* __restrict__ c3,
                            const float* __restrict__ grid,
                            const int*   __restrict__ coords,
                            const float* __restrict__ ln_g,
                            const float* __restrict__ ln_b,
                            const float* __restrict__ Wg_w,
                            const float* __restrict__ Wg_b,
                            float* __restrict__ R,
                            float* __restrict__ gate,
                            int*   __restrict__ flat2,
                            int*   __restrict__ flat3) {
    int lane = threadIdx.x & 31;
    int wave = threadIdx.x >> 5;
    int row  = blockIdx.x * 8 + wave;
    int b = row >> 12;            // row / 4096

    // ---- layernorm + gate ----
    const float* hr = h + (size_t)row * D_;
    float hv[6];
    float sum = 0.f, sq = 0.f;
    #pragma unroll
    for (int j = 0; j < 6; ++j) {
        float v = hr[lane + 32 * j];
        hv[j] = v; sum += v; sq += v * v;
    }
    sum = wave_sum(sum);
    sq  = wave_sum(sq);
    float mu  = sum * (1.f / D_);
    float var = sq * (1.f / D_) - mu * mu;
    float rs  = rsqrtf(var + 1e-5f);
    float dot = 0.f;
    #pragma unroll
    for (int j = 0; j < 6; ++j) {
        int ch = lane + 32 * j;
        float hn = (hv[j] - mu) * rs * ln_g[ch] + ln_b[ch];
        dot += hn * Wg_w[ch];
    }
    dot = wave_sum(dot);
    float g = 1.f / (1.f + __expf(-(dot + Wg_b[0])));

    // ---- bilinear gather both levels ----
    float gx = grid[(size_t)row * 2 + 0];
    float gy = grid[(size_t)row * 2 + 1];
    float r2[6], r3[6];
    bilin6(c2, b, 128, gx, gy, lane, r2);
    bilin6(c3, b,  64, gx, gy, lane, r3);
    float* Rr = R + (size_t)row * KR;
    #pragma unroll
    for (int j = 0; j < 6; ++j) {
        int ch = lane + 32 * j;
        Rr[ch]      = r2[j];
        Rr[ch + D_] = r3[j];
    }

    if (lane == 0) {
        gate[row] = g;
        int cx = coords[(size_t)row * 2 + 0];
        int cy = coords[(size_t)row * 2 + 1];
        flat2[row] = (cy >> 2) * 128 + (cx >> 2);   // stride 4 -> 128x128
        flat3[row] = (cy >> 3) * 64  + (cx >> 3);   // stride 8 -> 64x64
    }
}

// ---------------------------------------------------------------------------
// fp32 WMMA: one wave owns a 16(M) x 32(N) strip = two 16x16 C tiles, so each
// 8-byte A-fragment load feeds two v_wmma_f32_16x16x4_f32 ops (halves A-panel
// re-reads vs one-tile-per-wave). B panels staged in LDS as two contiguous
// [KDIM][16] arrays: lanes 0-15 and 16-31 hit disjoint bank groups (32 floats
// apart), conflict-free. A-fragment per ISA layout: lane%16 = M row,
// K = kk + 2*(lane/16) + {0,1}.
// ---------------------------------------------------------------------------
template <int KDIM>
__device__ __forceinline__ void wmma_strip_f32(const float* __restrict__ A, int lda,
                                               const float* __restrict__ Bs0,
                                               const float* __restrict__ Bs1,
                                               int rowBase, int lane,
                                               v8f& c0, v8f& c1) {
    int laneLo = lane & 15, laneHi = lane >> 4;
    const float* ap = A + (size_t)(rowBase + laneLo) * lda + 2 * laneHi;
    #pragma unroll 4
    for (int kk = 0; kk < KDIM; kk += 4) {
        v2f a = *(const v2f*)(ap + kk);
        int bidx = (kk + 2 * laneHi) * 16 + laneLo;
        v2f b0, b1;
        b0.x = Bs0[bidx];      b0.y = Bs0[bidx + 16];
        b1.x = Bs1[bidx];      b1.y = Bs1[bidx + 16];
        c0 = __builtin_amdgcn_wmma_f32_16x16x4_f32(false, a, false, b0,
                                                   (short)0, c0, false, false);
        c1 = __builtin_amdgcn_wmma_f32_16x16x4_f32(false, a, false, b1,
                                                   (short)0, c1, false, false);
    }
}

// read GEMM: h_out = h_in + gate * (R @ Wr + Wr_b)
__global__ void gemm_read_kernel(const float* __restrict__ R,
                                 const float* __restrict__ Wr,
                                 const float* __restrict__ Wr_b,
                                 const float* __restrict__ h_in,
                                 const float* __restrict__ gate,
                                 float* __restrict__ h_out) {
    __shared__ float Bs[2][KR * 16];         // 48 KB
    int n0 = blockIdx.y * 32;
    for (int idx = threadIdx.x; idx < KR * 32; idx += 256) {
        int kk = idx >> 5, n = idx & 31;
        Bs[n >> 4][kk * 16 + (n & 15)] = Wr[kk * D_ + n0 + n];
    }
    __syncthreads();
    int lane = threadIdx.x & 31, wave = threadIdx.x >> 5;
    int rowBase = blockIdx.x * 128 + wave * 16;
    v8f c0 = {}, c1 = {};
    wmma_strip_f32<KR>(R, KR, Bs[0], Bs[1], rowBase, lane, c0, c1);
    int laneLo = lane & 15, laneHi = lane >> 4;
    int colA = n0 + laneLo, colB = colA + 16;
    float biasA = Wr_b[colA], biasB = Wr_b[colB];
    #pragma unroll
    for (int i = 0; i < 8; ++i) {
        int row = rowBase + i + 8 * laneHi;   // ISA C/D layout: M = vgpr + 8*(lane/16)
        float g = gate[row];
        const float* hi = h_in + (size_t)row * D_;
        float*       ho = h_out + (size_t)row * D_;
        ho[colA] = hi[colA] + g * (c0[i] + biasA);
        ho[colB] = hi[colB] + g * (c1[i] + biasB);
    }
}

// scatter GEMM: canvas[b, col, flat[row]] += (h' @ W + b)[row, col]
__global__ void gemm_scatter_kernel(const float* __restrict__ Hn,
                                    const float* __restrict__ Wm,
                                    const float* __restrict__ Wb,
                                    const int*   __restrict__ flat,
                                    float* __restrict__ canvas_out,
                                    int HW) {
    __shared__ float Bs[2][D_ * 16];         // 24 KB
    int n0 = blockIdx.y * 32;
    for (int idx = threadIdx.x; idx < D_ * 32; idx += 256) {
        int kk = idx >> 5, n = idx & 31;
        Bs[n >> 4][kk * 16 + (n & 15)] = Wm[kk * D_ + n0 + n];
    }
    __syncthreads();
    int lane = threadIdx.x & 31, wave = threadIdx.x >> 5;
    int rowBase = blockIdx.x * 128 + wave * 16;
    v8f c0 = {}, c1 = {};
    wmma_strip_f32<D_>(Hn, D_, Bs[0], Bs[1], rowBase, lane, c0, c1);
    int laneLo = lane & 15, laneHi = lane >> 4;
    int colA = n0 + laneLo, colB = colA + 16;
    float biasA = Wb[colA], biasB = Wb[colB];
    #pragma unroll
    for (int i = 0; i < 8; ++i) {
        int row = rowBase + i + 8 * laneHi;
        int b = row >> 12;
        int fi = flat[row];
        size_t cbase = (size_t)b * D_ * HW;
        atomicAdd(canvas_out + cbase + (size_t)colA * HW + fi, c0[i] + biasA);
        atomicAdd(canvas_out + cbase + (size_t)colB * HW + fi, c1[i] + biasB);
    }
}

extern "C" void kernel_launch(void* const* d_in, const int* in_sizes, int n_in,
                              void* d_out, int out_size, void* d_ws, size_t ws_size,
                              hipStream_t stream) {
    const float* h      = (const float*)d_in[0];
    const float* c2     = (const float*)d_in[1];
    const float* c3     = (const float*)d_in[2];
    const float* grid   = (const float*)d_in[3];
    const int*   coords = (const int*)  d_in[4];
    const float* ln_g   = (const float*)d_in[5];
    const float* ln_b   = (const float*)d_in[6];
    const float* Wr_w   = (const float*)d_in[7];
    const float* Wr_b   = (const float*)d_in[8];
    const float* Wg_w   = (const float*)d_in[9];
    const float* Wg_b   = (const float*)d_in[10];
    const float* W2_w   = (const float*)d_in[11];
    const float* W2_b   = (const float*)d_in[12];
    const float* W3_w   = (const float*)d_in[13];
    const float* W3_b   = (const float*)d_in[14];

    const size_t H_ELEMS  = (size_t)ROWS * D_;              // 6,291,456
    const size_t C2_ELEMS = (size_t)B_ * D_ * 128 * 128;    // 25,165,824
    const size_t C3_ELEMS = (size_t)B_ * D_ * 64 * 64;      // 6,291,456

    float* out    = (float*)d_out;
    float* h_out  = out;
    float* c2_out = out + H_ELEMS;
    float* c3_out = out + H_ELEMS + C2_ELEMS;

    char* ws = (char*)d_ws;
    float* R     = (float*)ws;  ws += (size_t)ROWS * KR * sizeof(float);  // 50.3 MB
    float* gate  = (float*)ws;  ws += (size_t)ROWS * sizeof(float);
    int*   flat2 = (int*)ws;    ws += (size_t)ROWS * sizeof(int);
    int*   flat3 = (int*)ws;

    // scatter targets start as copies of the input canvases
    hipMemcpyAsync(c2_out, c2, C2_ELEMS * sizeof(float), hipMemcpyDeviceToDevice, stream);
    hipMemcpyAsync(c3_out, c3, C3_ELEMS * sizeof(float), hipMemcpyDeviceToDevice, stream);

    prep_kernel<<<ROWS / 8, 256, 0, stream>>>(h, c2, c3, grid, coords, ln_g, ln_b,
                                              Wg_w, Wg_b, R, gate, flat2, flat3);

    dim3 gdim(ROWS / 128, D_ / 32);   // 256 x 6 workgroups, 8 wave32 strips each
    gemm_read_kernel<<<gdim, 256, 0, stream>>>(R, Wr_w, Wr_b, h, gate, h_out);
    gemm_scatter_kernel<<<gdim, 256, 0, stream>>>(h_out, W2_w, W2_b, flat2, c2_out, 128 * 128);
    gemm_scatter_kernel<<<gdim, 256, 0, stream>>>(h_out, W3_w, W3_b, flat3, c3_out, 64 * 64);
}